// DPhysics_14207751815427
// MI455X (gfx1250) — compile-verified
//
#include <hip/hip_runtime.h>
#include <math.h>

#ifndef __has_builtin
#define __has_builtin(x) 0
#endif

// ---------------- problem constants (match reference) ----------------
#define HH      256
#define NMAXI   65535          // H*H - 1
#define NPTS    1024
#define NTHREADS 512
#define PPT     2              // points per thread
#define NWAVES  (NTHREADS / 32)
#define WIN     80             // LDS window edge (cells)
#define W0      88             // window start index (covers coords [-4.0, 4.0))

#define DTC     0.01f
#define DMAXC   12.8f
#define RESC    0.1f
#define MASSC   40.0f
#define FGRAVZ  (-392.0f)      // -MASS*GRAV
#define OMMAX   7.0f

// ---------------- CDNA5 async global->LDS staging (guarded) ----------------
// Probe-verified: __builtin_amdgcn_global_load_async_to_lds_b32 exists with
// plain int* params: (int* gsrc, int* ldsdst, imm offset, imm cpol)
#if __has_builtin(__builtin_amdgcn_global_load_async_to_lds_b32)
#define HAVE_ASYNC 1
#define ASYNC_G2L(gp, lp)                                                    \
  __builtin_amdgcn_global_load_async_to_lds_b32(                             \
      (int*)(void*)(gp), (int*)(void*)(lp), 0, 0)
#else
#define HAVE_ASYNC 0
#define ASYNC_G2L(gp, lp) do { *(lp) = *(gp); } while (0)
#endif

__device__ __forceinline__ void wait_async_done() {
#if HAVE_ASYNC
#if __has_builtin(__builtin_amdgcn_s_wait_asynccnt)
  __builtin_amdgcn_s_wait_asynccnt(0);
#else
  asm volatile("s_wait_asynccnt 0" ::: "memory");
#endif
#endif
}

// ---------------- fast tanh (v_tanh_f32 if present) ----------------
__device__ __forceinline__ float fast_tanh(float x) {
#if __has_builtin(__builtin_amdgcn_tanhf)
  return __builtin_amdgcn_tanhf(x);
#elif __has_builtin(__builtin_amdgcn_tanh_f32)
  return __builtin_amdgcn_tanh_f32(x);
#else
  return tanhf(x);
#endif
}

// rk4 with a constant derivative, written exactly like the reference
__device__ __forceinline__ float rk4c(float x, float xd) {
  float k1 = DTC * xd;
  float k2 = DTC * (xd + 0.5f * k1);
  float k3 = DTC * (xd + 0.5f * k2);
  float k4 = DTC * (xd + k3);
  return x + (k1 + 2.0f * k2 + 2.0f * k3 + k4) / 6.0f;
}

__device__ __forceinline__ int clamp_idx(int v) {
  return v < 0 ? 0 : (v > NMAXI ? NMAXI : v);
}

extern "C" __global__ void __launch_bounds__(NTHREADS)
dphys_sim_kernel(const float* __restrict__ z_grid,
                 const float* __restrict__ stiffness,
                 const float* __restrict__ damping,
                 const float* __restrict__ friction,
                 const float* __restrict__ controls,   // (T,B,2)
                 const float* __restrict__ x_points0,  // (B,NPTS,3)
                 const int*   __restrict__ track_ids,  // (NPTS)
                 float* __restrict__ out,              // (T,B,3)
                 int T, int Bn)
{
  const int b    = blockIdx.x;
  const int tid  = threadIdx.x;
  const int lane = tid & 31;
  const int wid  = tid >> 5;

  extern __shared__ float smem[];
  // Interleaved window: per cell a float4 = (z, k, c, f)
  float4* cell  = (float4*)smem;                      // WIN*WIN float4
  float*  cellf = smem;                               // scalar view
  float4* red4  = (float4*)(smem + WIN * WIN * 4);    // NWAVES*2 float4

  const size_t gb = (size_t)b * HH * HH;
  const float* gz = z_grid    + gb;
  const float* gk = stiffness + gb;
  const float* gc = damping   + gb;
  const float* gf = friction  + gb;

  // ---- stage the hot window of all 4 field grids into LDS, interleaved ----
  for (int i = tid; i < WIN * WIN; i += NTHREADS) {
    int r  = i / WIN;
    int c  = i - r * WIN;
    int go = (W0 + r) * HH + (W0 + c);
    ASYNC_G2L(gz + go, cellf + 4 * i + 0);
    ASYNC_G2L(gk + go, cellf + 4 * i + 1);
    ASYNC_G2L(gc + go, cellf + 4 * i + 2);
    ASYNC_G2L(gf + go, cellf + 4 * i + 3);
  }
  wait_async_done();
  __syncthreads();

  // ---- per-thread point state (registers) ----
  float px[PPT][3], pv[PPT][3];
  int   ptrk[PPT];
#pragma unroll
  for (int j = 0; j < PPT; ++j) {
    int p = tid + j * NTHREADS;
    size_t o = ((size_t)b * NPTS + p) * 3;
    px[j][0] = x_points0[o + 0];
    px[j][1] = x_points0[o + 1];
    px[j][2] = x_points0[o + 2];
    pv[j][0] = 0.0f; pv[j][1] = 0.0f; pv[j][2] = 0.0f;
    ptrk[j]  = track_ids[p];
  }

  // ---- rigid-body state, replicated in every thread ----
  float X0 = 0.f, X1 = 0.f, X2 = 0.f;
  float V0 = 0.f, V1 = 0.f, V2 = 0.f;
  float O0 = 0.f, O1 = 0.f, O2 = 0.f;
  float R[9] = {1.f, 0.f, 0.f, 0.f, 1.f, 0.f, 0.f, 0.f, 1.f};

  for (int t = 0; t < T; ++t) {
    const float vcmd = controls[((size_t)t * Bn + b) * 2 + 0];
    const float wcmd = controls[((size_t)t * Bn + b) * 2 + 1];
    const float tv0 = vcmd - wcmd * 0.25f;   // ROBOT_LY/2 = 0.25
    const float tv1 = vcmd + wcmd * 0.25f;

    // thrust_dir = normalize(R[:,0])
    float tdx = R[0], tdy = R[3], tdz = R[6];
    {
      float tn = sqrtf(tdx * tdx + tdy * tdy + tdz * tdz);
      float ti = 1.0f / fmaxf(tn, 1e-6f);
      tdx *= ti; tdy *= ti; tdz *= ti;
    }

    float sFx = 0.f, sFy = 0.f, sFz = 0.f;   // sum(F_spring + F_fric)
    float sTx = 0.f, sTy = 0.f, sTz = 0.f;   // sum torque

#pragma unroll
    for (int j = 0; j < PPT; ++j) {
      const float x = px[j][0], y = px[j][1], zc = px[j][2];

      const float xs = (x + DMAXC) / RESC;
      const float ys = (y + DMAXC) / RESC;
      const float xif = floorf(xs), yif = floorf(ys);
      const int xi = (int)xif, yi = (int)yif;
      const float xf = xs - xif, yf = ys - yif;

      // A=(xi,yi) B=(xi+1,yi) C=(xi,yi+1) D=(xi+1,yi+1)  (row=xi, col=yi)
      float4 A4, B4, C4, D4;
      const bool inwin = (xi >= W0) & (xi <= W0 + WIN - 2) &
                         (yi >= W0) & (yi <= W0 + WIN - 2);
      if (inwin) {
        const int bi = (xi - W0) * WIN + (yi - W0);
        A4 = cell[bi];
        B4 = cell[bi + WIN];
        C4 = cell[bi + 1];
        D4 = cell[bi + WIN + 1];
      } else {
        const int i00 = clamp_idx(yi + HH * xi);
        const int i01 = clamp_idx(yi + HH * (xi + 1));
        const int i10 = clamp_idx(yi + 1 + HH * xi);
        const int i11 = clamp_idx(yi + 1 + HH * (xi + 1));
        A4 = make_float4(gz[i00], gk[i00], gc[i00], gf[i00]);
        B4 = make_float4(gz[i01], gk[i01], gc[i01], gf[i01]);
        C4 = make_float4(gz[i10], gk[i10], gc[i10], gf[i10]);
        D4 = make_float4(gz[i11], gk[i11], gc[i11], gf[i11]);
      }

      const float w00 = (1.f - xf) * (1.f - yf);
      const float w01 = xf * (1.f - yf);
      const float w10 = (1.f - xf) * yf;
      const float w11 = xf * yf;

      const float zp = A4.x * w00 + B4.x * w01 + C4.x * w10 + D4.x * w11;
      const float kk = A4.y * w00 + B4.y * w01 + C4.y * w10 + D4.y * w11;
      const float cc = A4.z * w00 + B4.z * w01 + C4.z * w10 + D4.z * w11;
      const float ff = A4.w * w00 + B4.w * w01 + C4.w * w10 + D4.w * w11;

      const float dzdx = ((B4.x - A4.x) * (1.f - yf) + (D4.x - C4.x) * yf) / RESC;
      const float dzdy = ((C4.x - A4.x) * (1.f - xf) + (D4.x - B4.x) * xf) / RESC;
      float nx = -dzdx, ny = -dzdy, nz = 1.0f;
      {
        float nn = sqrtf(nx * nx + ny * ny + nz * nz);
        float ni = 1.0f / fmaxf(nn, 1e-6f);
        nx *= ni; ny *= ni; nz *= ni;
      }

      const float dh = zc - zp;
      const bool on_grid = (x >= -DMAXC) & (x <= DMAXC) &
                           (y >= -DMAXC) & (y <= DMAXC);
      const float contact = ((dh <= 0.0f) & on_grid) ? 1.0f : 0.0f;

      const float xd_n = pv[j][0] * nx + pv[j][1] * ny + pv[j][2] * nz;
      const float coef = -(kk * dh + cc * xd_n) * contact * (1.0f / (float)NPTS);
      const float Fsx = coef * nx, Fsy = coef * ny, Fsz = coef * nz;
      const float Nmag = sqrtf(Fsx * Fsx + Fsy * Fsy + Fsz * Fsz);

      const float tv = (ptrk[j] == 0) ? tv0 : tv1;
      const float dvx = tv * tdx - pv[j][0];
      const float dvy = tv * tdy - pv[j][1];
      const float dvz = tv * tdz - pv[j][2];
      const float dvn = dvx * nx + dvy * ny + dvz * nz;
      const float fN = ff * Nmag;
      const float Ffx = fN * fast_tanh(dvx - dvn * nx);
      const float Ffy = fN * fast_tanh(dvy - dvn * ny);
      const float Ffz = fN * fast_tanh(dvz - dvn * nz);

      const float Ftx = Fsx + Ffx, Fty = Fsy + Ffy, Ftz = Fsz + Ffz;
      const float rx = x - X0, ry = y - X1, rz = zc - X2;

      sFx += Ftx; sFy += Fty; sFz += Ftz;
      sTx += ry * Ftz - rz * Fty;
      sTy += rz * Ftx - rx * Ftz;
      sTz += rx * Fty - ry * Ftx;

      // xd_pts_new = xd + omega x r   (old body state)
      const float pvnx = V0 + (O1 * rz - O2 * ry);
      const float pvny = V1 + (O2 * rx - O0 * rz);
      const float pvnz = V2 + (O0 * ry - O1 * rx);
      px[j][0] = rk4c(x,  pvnx);
      px[j][1] = rk4c(y,  pvny);
      px[j][2] = rk4c(zc, pvnz);
      pv[j][0] = pvnx; pv[j][1] = pvny; pv[j][2] = pvnz;
    }

    // ---- wave32 butterfly reduction of the 6 sums ----
#pragma unroll
    for (int m = 16; m >= 1; m >>= 1) {
      sFx += __shfl_xor(sFx, m, 32);
      sFy += __shfl_xor(sFy, m, 32);
      sFz += __shfl_xor(sFz, m, 32);
      sTx += __shfl_xor(sTx, m, 32);
      sTy += __shfl_xor(sTy, m, 32);
      sTz += __shfl_xor(sTz, m, 32);
    }
    if (lane == 0) {
      red4[wid * 2 + 0] = make_float4(sFx, sFy, sFz, sTx);
      red4[wid * 2 + 1] = make_float4(sTy, sTz, 0.f, 0.f);
    }
    __syncthreads();

    float Fx = 0.f, Fy = 0.f, Fz = 0.f, Tx = 0.f, Ty = 0.f, Tz = 0.f;
#pragma unroll
    for (int wv = 0; wv < NWAVES; ++wv) {
      const float4 a = red4[wv * 2 + 0];
      const float4 c = red4[wv * 2 + 1];
      Fx += a.x; Fy += a.y; Fz += a.z;
      Tx += a.w; Ty += c.x; Tz += c.y;
    }

    // ---- rigid-body integration (replicated; identical in all threads) ----
    const float xdd0 = Fx / MASSC;
    const float xdd1 = Fy / MASSC;
    const float xdd2 = (Fz + FGRAVZ) / MASSC;
    float od0 = Tx * 1.0f;
    float od1 = Ty * (1.0f / 3.5f);
    float od2 = Tz * 0.25f;
    od0 = fminf(fmaxf(od0, -OMMAX), OMMAX);
    od1 = fminf(fmaxf(od1, -OMMAX), OMMAX);
    od2 = fminf(fmaxf(od2, -OMMAX), OMMAX);

    V0 = rk4c(V0, xdd0); V1 = rk4c(V1, xdd1); V2 = rk4c(V2, xdd2);
    X0 = rk4c(X0, V0);   X1 = rk4c(X1, V1);   X2 = rk4c(X2, V2);
    O0 = rk4c(O0, od0);  O1 = rk4c(O1, od1);  O2 = rk4c(O2, od2);

    // R = R @ (I + K*sin(th*dt) + K^2*(1-cos(th*dt))), K = skew(omega/th)
    {
      const float th = sqrtf(O0 * O0 + O1 * O1 + O2 * O2);
      const float ti = 1.0f / fmaxf(th, 1e-6f);
      const float ux = O0 * ti, uy = O1 * ti, uz = O2 * ti;
      const float st = sinf(th * DTC);
      const float ct = 1.0f - cosf(th * DTC);
      const float M00 = 1.f - ct * (uy * uy + uz * uz);
      const float M01 = -st * uz + ct * ux * uy;
      const float M02 =  st * uy + ct * ux * uz;
      const float M10 =  st * uz + ct * ux * uy;
      const float M11 = 1.f - ct * (ux * ux + uz * uz);
      const float M12 = -st * ux + ct * uy * uz;
      const float M20 = -st * uy + ct * ux * uz;
      const float M21 =  st * ux + ct * uy * uz;
      const float M22 = 1.f - ct * (ux * ux + uy * uy);
#pragma unroll
      for (int r3 = 0; r3 < 3; ++r3) {
        const float a = R[3 * r3 + 0], bb = R[3 * r3 + 1], cd = R[3 * r3 + 2];
        R[3 * r3 + 0] = a * M00 + bb * M10 + cd * M20;
        R[3 * r3 + 1] = a * M01 + bb * M11 + cd * M21;
        R[3 * r3 + 2] = a * M02 + bb * M12 + cd * M22;
      }
    }

    if (tid == 0) {
      const size_t oo = ((size_t)t * Bn + b) * 3;
      out[oo + 0] = X0; out[oo + 1] = X1; out[oo + 2] = X2;
    }
    __syncthreads();   // protect `red4` before next step's writes
  }
}

extern "C" void kernel_launch(void* const* d_in, const int* in_sizes, int n_in,
                              void* d_out, int out_size, void* d_ws, size_t ws_size,
                              hipStream_t stream) {
  (void)n_in; (void)out_size; (void)d_ws; (void)ws_size;
  const float* z_grid    = (const float*)d_in[0];
  const float* stiffness = (const float*)d_in[1];
  const float* damping   = (const float*)d_in[2];
  const float* friction  = (const float*)d_in[3];
  const float* controls  = (const float*)d_in[4];
  const float* x_points0 = (const float*)d_in[5];
  const int*   track_ids = (const int*)d_in[6];
  float* out = (float*)d_out;

  const int B = in_sizes[0] / (HH * HH);            // 64
  const int T = in_sizes[4] / (B * 2);              // 64

  const size_t shmem = (size_t)(4 * WIN * WIN + NWAVES * 8) * sizeof(float);
  (void)hipFuncSetAttribute((const void*)dphys_sim_kernel,
                            hipFuncAttributeMaxDynamicSharedMemorySize,
                            (int)shmem);
  dphys_sim_kernel<<<B, NTHREADS, shmem, stream>>>(
      z_grid, stiffness, damping, friction, controls, x_points0, track_ids,
      out, T, B);
}